// AttnFathers_46591805227336
// MI455X (gfx1250) — compile-verified
//
#include <hip/hip_runtime.h>

// Problem constants (match the JAX reference)
#define B_      16
#define MAXLEN  256
#define FNUM    32
#define H_      1024
#define ROWS_PER_B (MAXLEN * FNUM)   // 8192 rows of length H per batch

typedef __attribute__((ext_vector_type(2))) float v2f;
typedef __attribute__((ext_vector_type(4))) float f4;
typedef __attribute__((ext_vector_type(8))) float v8f;

// ---------------------------------------------------------------------------
// Kernel 1: v[b][n] = sum_k hidden[b][k] * W[k][n]
// 16x1024x1024 f32 GEMM via V_WMMA_F32_16X16X4_F32 (full f32 precision).
// One wave per 16-column tile of v -> 64 blocks x 32 threads.
// A (16x4 f32): VGPR0 = {lanes0-15: K=0, lanes16-31: K=2}, VGPR1 = {K=1, K=3}
// B (4x16 f32): lanes0-15 hold K=0..1 (V0,V1), lanes16-31 hold K=2..3
// D (16x16 f32): VGPR r = {lanes0-15: M=r, lanes16-31: M=r+8}, N = lane&15
// ---------------------------------------------------------------------------
__global__ __launch_bounds__(32)
void hw_gemm_wmma(const float* __restrict__ hidden,   // [B_, H_]
                  const float* __restrict__ W,        // [H_, H_] row-major (k, n)
                  float* __restrict__ v) {            // [B_, H_]
  const int lane = threadIdx.x;        // 0..31, wave32
  const int half = lane >> 4;          // 0 or 1
  const int l    = lane & 15;
  const int n0   = blockIdx.x * 16;

  v8f acc = {};
  for (int k0 = 0; k0 < H_; k0 += 4) {
    const int ka = k0 + 2 * half;      // K pair handled by this lane half
    v2f a, b;
    a.x = hidden[l * H_ + ka + 0];
    a.y = hidden[l * H_ + ka + 1];
    b.x = W[(ka + 0) * H_ + n0 + l];
    b.y = W[(ka + 1) * H_ + n0 + l];
    // (neg_a, A, neg_b, B, c_mod, C, reuse_a, reuse_b)
    acc = __builtin_amdgcn_wmma_f32_16x16x4_f32(false, a, false, b,
                                                (short)0, acc, false, false);
  }
#pragma unroll
  for (int r = 0; r < 8; ++r) {
    const int brow = r + 8 * half;     // batch index (M)
    v[brow * H_ + n0 + l] = acc[r];
  }
}

// ---------------------------------------------------------------------------
// Kernel 2: st[b][f][m] = dot(FO[b,m,f,:], v[b,:])
// Pure-bandwidth kernel: streams 512 MB once. Non-temporal b128 loads,
// v[b] cached in registers (32 floats/lane), wave32 shuffle reduction.
// 256 threads = 8 waves per block; each wave does 8 rows -> 64 rows/block.
// ---------------------------------------------------------------------------
__global__ __launch_bounds__(256)
void scores_kernel(const f4* __restrict__ fo,     // [B_*ROWS_PER_B, H_/4]
                   const float* __restrict__ v,   // [B_, H_]
                   float* __restrict__ st) {      // [B_, FNUM, MAXLEN]
  const int lane  = threadIdx.x & 31;
  const int wid   = threadIdx.x >> 5;
  const int b     = blockIdx.x >> 7;              // 128 blocks per batch
  const int chunk = blockIdx.x & 127;
  const int rowBase = chunk * 64 + wid * 8;

  // Each lane keeps its 32 v-values (8 float4, strided by 128 floats)
  const f4* vq = (const f4*)(v + b * H_);
  f4 vv[8];
#pragma unroll
  for (int j = 0; j < 8; ++j) vv[j] = vq[j * 32 + lane];

#pragma unroll
  for (int i = 0; i < 8; ++i) {
    const int R = rowBase + i;                    // R = m*FNUM + f
    const f4* p = fo + (size_t)(b * ROWS_PER_B + R) * (H_ / 4);
    float acc = 0.f;
#pragma unroll
    for (int j = 0; j < 8; ++j) {
      f4 x = __builtin_nontemporal_load(&p[j * 32 + lane]);
      acc += x.x * vv[j].x + x.y * vv[j].y + x.z * vv[j].z + x.w * vv[j].w;
    }
#pragma unroll
    for (int m = 16; m >= 1; m >>= 1) acc += __shfl_xor(acc, m, 32);
    if (lane == 0) {
      const int mrow = R >> 5;                    // R / FNUM
      const int f    = R & 31;                    // R % FNUM
      st[b * ROWS_PER_B + f * MAXLEN + mrow] = acc;
    }
  }
}

// ---------------------------------------------------------------------------
// Kernel 3: out[b][m][f] = softmax_m(st[b][f][m])
// One wave per (b,f) pair; 8 contiguous m-values per lane. The bias term
// hidden.attn_b is constant over m and cancels in the softmax -> skipped.
// ---------------------------------------------------------------------------
__global__ __launch_bounds__(256)
void softmax_kernel(const float* __restrict__ st,  // [B_, FNUM, MAXLEN]
                    float* __restrict__ out) {     // [B_, MAXLEN, FNUM]
  const int lane = threadIdx.x & 31;
  const int wid  = threadIdx.x >> 5;
  const int idx  = blockIdx.x * 8 + wid;           // 0..511
  const int b    = idx >> 5;
  const int f    = idx & 31;

  const float* p = st + b * ROWS_PER_B + f * MAXLEN;
  float x[8];
#pragma unroll
  for (int t = 0; t < 8; ++t) x[t] = p[lane * 8 + t];

  float mx = x[0];
#pragma unroll
  for (int t = 1; t < 8; ++t) mx = fmaxf(mx, x[t]);
#pragma unroll
  for (int m = 16; m >= 1; m >>= 1) mx = fmaxf(mx, __shfl_xor(mx, m, 32));

  float s = 0.f;
#pragma unroll
  for (int t = 0; t < 8; ++t) { x[t] = __expf(x[t] - mx); s += x[t]; }
#pragma unroll
  for (int m = 16; m >= 1; m >>= 1) s += __shfl_xor(s, m, 32);

  const float inv = 1.0f / s;
#pragma unroll
  for (int t = 0; t < 8; ++t)
    out[b * ROWS_PER_B + (lane * 8 + t) * FNUM + f] = x[t] * inv;
}

// ---------------------------------------------------------------------------
// Host launcher
// Inputs (setup_inputs order): hidden[1,B,H] f32, fathers_outputs[B,M,F,H] f32,
// fathers_lengths[B] i32 (unused by reference), attn_W[H,H] f32,
// attn_b[H] f32 (cancels in softmax -> unused).
// ---------------------------------------------------------------------------
extern "C" void kernel_launch(void* const* d_in, const int* in_sizes, int n_in,
                              void* d_out, int out_size, void* d_ws, size_t ws_size,
                              hipStream_t stream) {
  (void)in_sizes; (void)n_in; (void)out_size; (void)ws_size;

  const float* hidden = (const float*)d_in[0];
  const float* fo     = (const float*)d_in[1];
  // d_in[2] = fathers_lengths: not used by the reference computation
  const float* attn_W = (const float*)d_in[3];
  // d_in[4] = attn_b: adds a per-b constant to the logits -> cancels in softmax

  float* out = (float*)d_out;

  // Workspace layout: v[B_*H_] (64 KB) then st[B_*FNUM*MAXLEN] (512 KB)
  float* v  = (float*)d_ws;
  float* st = v + (size_t)B_ * H_;

  hw_gemm_wmma<<<H_ / 16, 32, 0, stream>>>(hidden, attn_W, v);
  scores_kernel<<<B_ * 128, 256, 0, stream>>>((const f4*)fo, v, st);
  softmax_kernel<<<(B_ * FNUM) / 8, 256, 0, stream>>>(st, out);
}